// EnergyAE_49778670961138
// MI455X (gfx1250) — compile-verified
//
#include <hip/hip_runtime.h>
#include <hip/hip_bf16.h>

// ============================================================================
// EnergyAE for MI455X (gfx1250): fp32 WMMA GEMMs + Tensor Data Mover staging.
//
//   K      = dec_W2 @ dec_W2^T               (H x H, 103 GFLOP — flagship)
//   A_b    = dec_W1 masked by relu'(z@dec_W1+b1)
//   G_b    = A_b K A_b^T ;  t_b = A_b (dec_W2 @ delta_b)
//   eigvalsh avoided: Prec = L L^T => logdet/2 = sum log L_ii,
//                     sum 1/eig = ||L^{-1}||_F^2
//
// GEMM data path: tensor_load_to_lds (TDM, TENSORcnt) double-buffered — wave 0
// issues the DMA for chunk i+1, all waves run v_wmma_f32_16x16x4_f32 on chunk i.
// ============================================================================

#define B_ 64
#define D_ 12288
#define N_ 32
#define H_ 2048

typedef __attribute__((ext_vector_type(2))) float v2f;
typedef __attribute__((ext_vector_type(8))) float v8f;
typedef unsigned int u32x4 __attribute__((ext_vector_type(4)));
typedef int i32x8 __attribute__((ext_vector_type(8)));
typedef int i32x4 __attribute__((ext_vector_type(4)));

#define KC 32          // k-chunk staged in LDS
#define LDA_T (KC + 4) // padded LDS row stride for k-major tiles (36 floats)

// LDS byte offset of a generic pointer to __shared__ (low 32 bits of flat addr)
__device__ __forceinline__ unsigned lds_off(const void* p)
{
    return (unsigned)(unsigned long long)(uintptr_t)p;
}

// ----------------------------------------------------------------------------
// TDM 2-D tile load: global (tile_y rows x tile_x dwords, row stride
// stride_elems dwords) -> LDS at lds_byte, inserting (pad_m1+1) dwords of
// padding after every tile_x dwords (tile_x must be 2^(piv+1)).
// D# built per CDNA5 ISA; tracked by TENSORcnt.
// Toolchain here exposes the 6-arg builtin:
//   (u32x4 g0, i32x8 g1, i32x4 g2, i32x4 g3, i32x8 extra, i32 cpol)
// ----------------------------------------------------------------------------
__device__ __forceinline__ void tdm_load_2d(unsigned lds_byte, const void* gptr,
                                            unsigned tile_x, unsigned tile_y,
                                            unsigned stride_elems,
                                            unsigned piv, unsigned pad_m1)
{
    const unsigned long long ga = (unsigned long long)(uintptr_t)gptr;
    u32x4 g0;
    g0[0] = 1u;                                   // count=1, user descriptor
    g0[1] = lds_byte;                             // lds_addr
    g0[2] = (unsigned)ga;                         // global_addr[31:0]
    g0[3] = (unsigned)(ga >> 32) | (2u << 30);    // global_addr[56:32] | type=2
    i32x8 g1;
    g1[0] = (int)((2u << 16) | (1u << 20) | (piv << 22) | (pad_m1 << 25));
    g1[1] = (int)((tile_x & 0xFFFFu) << 16);      // tensor_dim0[15:0]
    g1[2] = (int)((tile_x >> 16) | ((tile_y & 0xFFFFu) << 16)); // td0 hi | td1 lo
    g1[3] = (int)((tile_y >> 16) | (tile_x << 16));             // td1 hi | tile_dim0
    g1[4] = (int)(tile_y & 0xFFFFu);              // tile_dim1 (tile_dim2=0)
    g1[5] = (int)stride_elems;                    // tensor_dim0_stride[31:0]
    g1[6] = 0;
    g1[7] = 0;
    const i32x4 z4 = {0, 0, 0, 0};                // 2-D: groups 2/3 unused
    const i32x8 z8 = {0, 0, 0, 0, 0, 0, 0, 0};
    __builtin_amdgcn_tensor_load_to_lds(g0, g1, z4, z4, z8, 0);
}

// ----------------------------------------------------------------------------
// fp32 WMMA GEMM:  C[b] = epilogue( A[b] @ op(B[b]) )
//   A: M x Kd row-major.  BT=1: B is Nd x Kd row-major (A @ B^T), tiles staged
//   n-major/k-contiguous (b64 fragments). BT=0: B is Kd x Nd row-major, tiles
//   staged k-major/n-contiguous (2x b32 fragments).
//   Double-buffered TDM staging; 4 waves, each owns 32x32 = 2x2 WMMA tiles.
// Requires: M%BM==0, Nd%BN==0, Kd%KC==0, BN in {64,128}.
// ----------------------------------------------------------------------------
template <int BM, int BN, int BT>
__global__ __launch_bounds__(128) void gemm_f32_wmma(
    const float* __restrict__ A, long long sA, int lda,
    const float* __restrict__ B, long long sB, int ldb,
    const float* __restrict__ bias,
    const float* __restrict__ sub, long long sSub,
    float* __restrict__ C, long long sC, int ldc,
    int Kd, int relu)
{
    static_assert((BM / 32) * (BN / 32) == 4, "4 waves per block");
    constexpr int BROW = BT ? LDA_T : (BN + 4);            // LDS row stride (B)
    constexpr int BSZ  = BT ? BN * LDA_T : KC * (BN + 4);  // floats per B buffer
    constexpr unsigned BPIV = (BN == 64) ? 5u : 6u;        // pad interval: BN dwords
    __shared__ alignas(16) float As[2 * BM * LDA_T];
    __shared__ alignas(16) float Bs[2 * BSZ];

    const int tid = threadIdx.x;
    const int bn = blockIdx.x * BN;
    const int bm = blockIdx.y * BM;
    const long long bz = blockIdx.z;
    A += bz * sA;
    B += bz * sB;
    C += bz * sC;
    if (sub) sub += bz * sSub;

    const int wave = tid >> 5;
    const int lane = tid & 31;
    const int half = lane >> 4;
    const int l16  = lane & 15;
    constexpr int WN = BN / 32;
    const int wm = (wave / WN) * 32;
    const int wn = (wave % WN) * 32;

    const float* Arow = A + (long long)bm * lda;   // tile row start, advance by k0
    const float* Brow = BT ? (B + (long long)bn * ldb) : (B + bn);

    v8f acc[2][2] = {};
    const int nch = Kd / KC;

    // prolog: DMA chunk 0 into buffer 0 (wave 0 only; EXEC ignored by TDM)
    if (wave == 0) {
        tdm_load_2d(lds_off(&As[0]), Arow, KC, BM, (unsigned)lda, 4u, 3u);
        if (BT)
            tdm_load_2d(lds_off(&Bs[0]), Brow, KC, BN, (unsigned)ldb, 4u, 3u);
        else
            tdm_load_2d(lds_off(&Bs[0]), Brow, BN, KC, (unsigned)ldb, BPIV, 3u);
    }

    for (int i = 0; i < nch; ++i) {
        const int buf = i & 1;
        if (wave == 0) __builtin_amdgcn_s_wait_tensorcnt(0);
        __syncthreads();   // chunk i resident; prior compute finished everywhere

        // kick off DMA for chunk i+1 into the other buffer while we compute
        if (wave == 0 && (i + 1 < nch)) {
            const int k1 = (i + 1) * KC;
            tdm_load_2d(lds_off(&As[(1 - buf) * BM * LDA_T]), Arow + k1,
                        KC, BM, (unsigned)lda, 4u, 3u);
            if (BT)
                tdm_load_2d(lds_off(&Bs[(1 - buf) * BSZ]), Brow + k1,
                            KC, BN, (unsigned)ldb, 4u, 3u);
            else
                tdm_load_2d(lds_off(&Bs[(1 - buf) * BSZ]),
                            Brow + (long long)k1 * ldb,
                            BN, KC, (unsigned)ldb, BPIV, 3u);
        }

        const float* Al = &As[buf * BM * LDA_T];
        const float* Bl = &Bs[buf * BSZ];
        // f32 16x16x4 fragments: A lanes0-15 K={k,k+1}, lanes16-31 K={k+2,k+3};
        // B mirrors with N striped over l16. C/D: VGPR g -> M=g+8*half, N=l16.
#pragma unroll
        for (int k = 0; k < KC; k += 4) {
#pragma unroll
            for (int tm = 0; tm < 2; ++tm) {
                const v2f av =
                    *(const v2f*)(&Al[(wm + tm * 16 + l16) * LDA_T + k + 2 * half]);
#pragma unroll
                for (int tn = 0; tn < 2; ++tn) {
                    v2f bv;
                    if (BT) {
                        bv = *(const v2f*)(&Bl[(wn + tn * 16 + l16) * BROW + k + 2 * half]);
                    } else {
                        const int col = wn + tn * 16 + l16;
                        bv.x = Bl[(k + 2 * half) * BROW + col];
                        bv.y = Bl[(k + 2 * half + 1) * BROW + col];
                    }
                    acc[tm][tn] = __builtin_amdgcn_wmma_f32_16x16x4_f32(
                        false, av, false, bv, (short)0, acc[tm][tn], false, false);
                }
            }
        }
        __syncthreads();   // everyone done reading buf before DMA i+2 overwrites
    }

#pragma unroll
    for (int tm = 0; tm < 2; ++tm) {
#pragma unroll
        for (int tn = 0; tn < 2; ++tn) {
            const int col = bn + wn + tn * 16 + l16;
#pragma unroll
            for (int g = 0; g < 8; ++g) {
                const int row = bm + wm + tm * 16 + g + 8 * half;
                float v = acc[tm][tn][g];
                if (bias) v += bias[col];
                if (relu) v = v > 0.f ? v : 0.f;
                const long long idx = (long long)row * ldc + col;
                if (sub) v = sub[idx] - v;
                C[idx] = v;
            }
        }
    }
}

// ----------------------------------------------------------------------------
// z_star = H1 @ enc_W2 + enc_b2 ; sp/sv = exp(z_star@sig_W+sig_b)
// ----------------------------------------------------------------------------
__global__ __launch_bounds__(256) void zstar_kernel(
    const float* __restrict__ H1, const float* __restrict__ W2,
    const float* __restrict__ b2, const float* __restrict__ sigW,
    const float* __restrict__ sigb,
    float* __restrict__ z, float* __restrict__ sp, float* __restrict__ sv)
{
    const int b = blockIdx.x;
    const int n = threadIdx.x & 31, s = threadIdx.x >> 5;
    const float* h = H1 + (long long)b * H_;
    float acc = 0.f;
    for (int i = 0; i < H_ / 8; ++i) {
        const int hh = s * (H_ / 8) + i;
        acc += h[hh] * W2[hh * N_ + n];
    }
    __shared__ float red[8][N_];
    red[s][n] = acc;
    __syncthreads();
    if (s == 0) {
        float z0 = b2[n];
        for (int j = 0; j < 8; ++j) z0 += red[j][n];
        z[b * N_ + n] = z0;
        red[0][n] = z0;
    }
    __syncthreads();
    if (threadIdx.x == 0) {
        float a0 = sigb[0], a1 = sigb[1];
        for (int j = 0; j < N_; ++j) {
            a0 += red[0][j] * sigW[2 * j];
            a1 += red[0][j] * sigW[2 * j + 1];
        }
        sp[b] = __expf(a0);
        sv[b] = __expf(a1);
    }
}

// ----------------------------------------------------------------------------
// u = z@dec_W1 + b1; A[b][n][h] = dec_W1[n][h]*(u>0); optional hout = relu(u).
// ----------------------------------------------------------------------------
__global__ __launch_bounds__(256) void build_A_kernel(
    const float* __restrict__ z, const float* __restrict__ W1,
    const float* __restrict__ b1, float* __restrict__ Aout,
    float* __restrict__ hout)
{
    const int b = blockIdx.y;
    const int h = blockIdx.x * 256 + threadIdx.x;
    __shared__ float zs[N_];
    if (threadIdx.x < N_) zs[threadIdx.x] = z[b * N_ + threadIdx.x];
    __syncthreads();
    float u = b1[h];
#pragma unroll 8
    for (int n = 0; n < N_; ++n) u += zs[n] * W1[n * H_ + h];
    const float m = u > 0.f ? 1.f : 0.f;
    if (hout) hout[(long long)b * H_ + h] = u > 0.f ? u : 0.f;
    float* Ab = Aout + (long long)b * N_ * H_ + h;
#pragma unroll 8
    for (int n = 0; n < N_; ++n) Ab[n * H_] = m * W1[n * H_ + h];
}

// ----------------------------------------------------------------------------
// G[b] = T[b] @ A[b]^T (32x32); optionally Prec = G/sp^2 + sig_term + I.
// ----------------------------------------------------------------------------
#define GLD 133
__global__ __launch_bounds__(1024) void g_prec_kernel(
    const float* __restrict__ T, const float* __restrict__ A,
    const float* __restrict__ sp, const float* __restrict__ sigW,
    float* __restrict__ out, int makePrec)
{
    const int b = blockIdx.x;
    const int tid = threadIdx.x;
    const int n = tid >> 5, m = tid & 31;
    const float* Tb = T + (long long)b * N_ * H_;
    const float* Ab = A + (long long)b * N_ * H_;
    __shared__ float Ts[N_ * GLD];
    __shared__ float Asd[N_ * GLD];
    float acc = 0.f;
    for (int k0 = 0; k0 < H_; k0 += 128) {
        for (int i = tid; i < N_ * 128; i += 1024) {
            const int r = i >> 7, c = i & 127;
            Ts[r * GLD + c]  = Tb[(long long)r * H_ + k0 + c];
            Asd[r * GLD + c] = Ab[(long long)r * H_ + k0 + c];
        }
        __syncthreads();
#pragma unroll 8
        for (int k = 0; k < 128; ++k)
            acc += Ts[n * GLD + k] * Asd[m * GLD + k];
        __syncthreads();
    }
    float v = acc;
    if (makePrec) {
        const float s = sp[b];
        const float st = 0.5f * ((float)N_ * sigW[2 * n] * sigW[2 * m] +
                                 (float)(D_ - N_) * sigW[2 * n + 1] * sigW[2 * m + 1]);
        v = acc / (s * s) + st + (n == m ? 1.f : 0.f);
    }
    out[(long long)b * N_ * N_ + n * N_ + m] = v;
}

// ----------------------------------------------------------------------------
// Per-batch (one wave): Prec = L L^T -> logdet, trace(Prec^{-1}),
// dz = L^{-T} eps, z_s = z_star + dz, sp2/sv2 = exp(z_s@sigW+sigb).
// ----------------------------------------------------------------------------
__global__ __launch_bounds__(32) void chol_energy_kernel(
    const float* __restrict__ Prec, const float* __restrict__ eps,
    const float* __restrict__ zstar, const float* __restrict__ sigW,
    const float* __restrict__ sigb,
    float* __restrict__ z_s, float* __restrict__ logdet,
    float* __restrict__ trinv, float* __restrict__ sp2, float* __restrict__ sv2)
{
    const int b = blockIdx.x, t = threadIdx.x;
    __shared__ float L[N_][N_ + 1];
    __shared__ float X[N_][N_ + 1];
    __shared__ float r[N_], zsv[N_], red[N_];
    for (int i = 0; i < N_; ++i) L[i][t] = Prec[(long long)b * N_ * N_ + i * N_ + t];
    __syncthreads();
    for (int k = 0; k < N_; ++k) {
        if (t == k) L[k][k] = sqrtf(L[k][k]);
        __syncthreads();
        if (t > k) L[t][k] = L[t][k] / L[k][k];
        __syncthreads();
        if (t > k) {
            const float lik = L[t][k];
            for (int j = k + 1; j <= t; ++j) L[t][j] -= lik * L[j][k];
        }
        __syncthreads();
    }
    r[t] = eps[b * N_ + t];
    __syncthreads();
    for (int i = N_ - 1; i >= 0; --i) {
        if (t == i) r[i] = r[i] / L[i][i];
        __syncthreads();
        if (t < i) r[t] -= L[i][t] * r[i];
        __syncthreads();
    }
    const float zsval = zstar[b * N_ + t] + r[t];
    z_s[b * N_ + t] = zsval;
    zsv[t] = zsval;
    for (int i = t; i < N_; ++i) {           // lane-private column of L^{-1}
        float s = (i == t) ? 1.f : 0.f;
        for (int j = t; j < i; ++j) s -= L[i][j] * X[j][t];
        X[i][t] = s / L[i][i];
    }
    float f = 0.f;
    for (int i = t; i < N_; ++i) f += X[i][t] * X[i][t];
    red[t] = f;
    __syncthreads();
    if (t == 0) {
        float tr = 0.f, ld = 0.f, a0 = sigb[0], a1 = sigb[1];
        for (int j = 0; j < N_; ++j) {
            tr += red[j];
            ld += logf(L[j][j]);
            a0 += zsv[j] * sigW[2 * j];
            a1 += zsv[j] * sigW[2 * j + 1];
        }
        trinv[b] = tr;
        logdet[b] = ld;
        sp2[b] = __expf(a0);
        sv2[b] = __expf(a1);
    }
}

__global__ __launch_bounds__(256) void dsq_kernel(const float* __restrict__ delta,
                                                  float* __restrict__ dsq)
{
    const int b = blockIdx.x, t = threadIdx.x;
    const float* db = delta + (long long)b * D_;
    float a = 0.f;
    for (int i = t; i < D_; i += 256) { const float d = db[i]; a += d * d; }
    __shared__ float red[256];
    red[t] = a;
    __syncthreads();
    for (int s = 128; s > 0; s >>= 1) {
        if (t < s) red[t] += red[t + s];
        __syncthreads();
    }
    if (t == 0) dsq[b] = red[0];
}

__global__ __launch_bounds__(256) void tvec_kernel(const float* __restrict__ A2,
                                                   const float* __restrict__ V,
                                                   float* __restrict__ tvec)
{
    const int b = blockIdx.x, tid = threadIdx.x;
    const int n = tid >> 3, sl = tid & 7;
    const float* An = A2 + (long long)b * N_ * H_ + (long long)n * H_;
    const float* Vb = V + (long long)b * H_;
    float a = 0.f;
    for (int i = 0; i < H_ / 8; ++i) { const int h = i * 8 + sl; a += An[h] * Vb[h]; }
    __shared__ float red[N_][8];
    red[n][sl] = a;
    __syncthreads();
    if (sl == 0) {
        float s = 0.f;
        for (int j = 0; j < 8; ++j) s += red[n][j];
        tvec[b * N_ + n] = s;
    }
}

// ----------------------------------------------------------------------------
// Final per-batch: chol(G2), solve G2 sol = t, d_proj = t.sol, assemble output.
// ----------------------------------------------------------------------------
__global__ __launch_bounds__(32) void final_kernel(
    const float* __restrict__ G2, const float* __restrict__ tvec,
    const float* __restrict__ dsq, const float* __restrict__ sp2,
    const float* __restrict__ sv2, const float* __restrict__ logdet,
    const float* __restrict__ trinv, const float* __restrict__ zstar,
    float* __restrict__ outp)
{
    const int b = blockIdx.x, t = threadIdx.x;
    __shared__ float L[N_][N_ + 1];
    __shared__ float y[N_], red[N_], red2[N_];
    for (int i = 0; i < N_; ++i) L[i][t] = G2[(long long)b * N_ * N_ + i * N_ + t];
    __syncthreads();
    for (int k = 0; k < N_; ++k) {
        if (t == k) L[k][k] = sqrtf(L[k][k]);
        __syncthreads();
        if (t > k) L[t][k] = L[t][k] / L[k][k];
        __syncthreads();
        if (t > k) {
            const float lik = L[t][k];
            for (int j = k + 1; j <= t; ++j) L[t][j] -= lik * L[j][k];
        }
        __syncthreads();
    }
    y[t] = tvec[b * N_ + t];
    __syncthreads();
    for (int i = 0; i < N_; ++i) {
        if (t == i) y[i] = y[i] / L[i][i];
        __syncthreads();
        if (t > i) y[t] -= L[t][i] * y[i];
        __syncthreads();
    }
    for (int i = N_ - 1; i >= 0; --i) {
        if (t == i) y[i] = y[i] / L[i][i];
        __syncthreads();
        if (t < i) y[t] -= L[i][t] * y[i];
        __syncthreads();
    }
    const float zz = zstar[b * N_ + t];
    red[t] = tvec[b * N_ + t] * y[t];
    red2[t] = zz * zz;
    __syncthreads();
    if (t == 0) {
        float dp = 0.f, zn = 0.f;
        for (int j = 0; j < N_; ++j) { dp += red[j]; zn += red2[j]; }
        const float spb = sp2[b], svb = sv2[b];
        const float i2sp = 0.5f / (spb * spb), i2sv = 0.5f / (svb * svb);
        const float recon = dsq[b] * i2sv + dp * (i2sp - i2sv) +
                            (float)N_ * logf(spb) + (float)(D_ - N_) * logf(svb);
        const float latent = zn * 0.5f + trinv[b] * 0.5f;
        outp[b] = (recon + latent + logdet[b]) / (float)D_;
    }
}

// ============================================================================
extern "C" void kernel_launch(void* const* d_in, const int* in_sizes, int n_in,
                              void* d_out, int out_size, void* d_ws, size_t ws_size,
                              hipStream_t stream)
{
    (void)in_sizes; (void)n_in; (void)out_size; (void)ws_size;
    const float* x     = (const float*)d_in[0];
    const float* eps   = (const float*)d_in[1];
    const float* encW1 = (const float*)d_in[2];
    const float* encb1 = (const float*)d_in[3];
    const float* encW2 = (const float*)d_in[4];
    const float* encb2 = (const float*)d_in[5];
    const float* decW1 = (const float*)d_in[6];
    const float* decb1 = (const float*)d_in[7];
    const float* decW2 = (const float*)d_in[8];
    const float* decb2 = (const float*)d_in[9];
    const float* sigW  = (const float*)d_in[10];
    const float* sigb  = (const float*)d_in[11];
    float* out = (float*)d_out;

    float* w = (float*)d_ws;
    size_t o = 0;
    float* H1    = w + o; o += (size_t)B_ * H_;
    float* ZST   = w + o; o += (size_t)B_ * N_;
    float* ZS    = w + o; o += (size_t)B_ * N_;
    float* SP    = w + o; o += B_;
    float* SV    = w + o; o += B_;  (void)SV;
    float* SP2   = w + o; o += B_;
    float* SV2   = w + o; o += B_;
    float* DSQ   = w + o; o += B_;
    float* LOGD  = w + o; o += B_;
    float* TRIN  = w + o; o += B_;
    o = (o + 63) & ~(size_t)63;
    float* TVEC  = w + o; o += (size_t)B_ * N_;
    float* KMAT  = w + o; o += (size_t)H_ * H_;
    float* ABUF  = w + o; o += (size_t)B_ * N_ * H_;
    float* TBUF  = w + o; o += (size_t)B_ * N_ * H_;
    float* PREC  = w + o; o += (size_t)B_ * N_ * N_;
    float* DELTA = w + o; o += (size_t)B_ * D_;
    float* VBUF  = w + o; o += (size_t)B_ * H_;

    const long long sNH = (long long)N_ * H_;

    // 1) H1 = relu(x @ enc_W1 + b1)
    gemm_f32_wmma<64, 64, 0><<<dim3(H_ / 64, B_ / 64, 1), 128, 0, stream>>>(
        x, 0, D_, encW1, 0, H_, encb1, nullptr, 0, H1, 0, H_, D_, 1);
    // 2) z_star, sp, sv
    zstar_kernel<<<B_, 256, 0, stream>>>(H1, encW2, encb2, sigW, sigb, ZST, SP, SV);
    // 3) K = dec_W2 @ dec_W2^T
    gemm_f32_wmma<64, 64, 1><<<dim3(H_ / 64, H_ / 64, 1), 128, 0, stream>>>(
        decW2, 0, D_, decW2, 0, D_, nullptr, nullptr, 0, KMAT, 0, H_, D_, 0);
    // 4) A1 = masked dec_W1 at z_star
    build_A_kernel<<<dim3(H_ / 256, B_), 256, 0, stream>>>(ZST, decW1, decb1, ABUF, nullptr);
    // 5) T = A1 @ K (batched)
    gemm_f32_wmma<32, 128, 0><<<dim3(H_ / 128, 1, B_), 128, 0, stream>>>(
        ABUF, sNH, H_, KMAT, 0, H_, nullptr, nullptr, 0, TBUF, sNH, H_, H_, 0);
    // 6) Prec = (T A1^T)/sp^2 + sig_term + I
    g_prec_kernel<<<B_, 1024, 0, stream>>>(TBUF, ABUF, SP, sigW, PREC, 1);
    // 7) Cholesky energy terms + z_s
    chol_energy_kernel<<<B_, 32, 0, stream>>>(PREC, eps, ZST, sigW, sigb,
                                              ZS, LOGD, TRIN, SP2, SV2);
    // 8) A2 + h2 at z_s (reuse H1)
    build_A_kernel<<<dim3(H_ / 256, B_), 256, 0, stream>>>(ZS, decW1, decb1, ABUF, H1);
    // 9) delta = x - (h2 @ dec_W2 + b2)
    gemm_f32_wmma<64, 64, 0><<<dim3(D_ / 64, B_ / 64, 1), 128, 0, stream>>>(
        H1, 0, H_, decW2, 0, D_, decb2, x, 0, DELTA, 0, D_, H_, 0);
    // 10) d_sq
    dsq_kernel<<<B_, 256, 0, stream>>>(DELTA, DSQ);
    // 11) V = delta @ dec_W2^T
    gemm_f32_wmma<64, 64, 1><<<dim3(H_ / 64, B_ / 64, 1), 128, 0, stream>>>(
        DELTA, 0, D_, decW2, 0, D_, nullptr, nullptr, 0, VBUF, 0, H_, D_, 0);
    // 12) t = A2 @ V
    tvec_kernel<<<B_, 256, 0, stream>>>(ABUF, VBUF, TVEC);
    // 13) T = A2 @ K ; 14) G2 = T A2^T
    gemm_f32_wmma<32, 128, 0><<<dim3(H_ / 128, 1, B_), 128, 0, stream>>>(
        ABUF, sNH, H_, KMAT, 0, H_, nullptr, nullptr, 0, TBUF, sNH, H_, H_, 0);
    g_prec_kernel<<<B_, 1024, 0, stream>>>(TBUF, ABUF, SP, sigW, PREC, 0);
    // 15) final assembly
    final_kernel<<<B_, 32, 0, stream>>>(PREC, TVEC, DSQ, SP2, SV2, LOGD, TRIN, ZST, out);
}